// KVQuantWrapper_5652176962322
// MI455X (gfx1250) — compile-verified
//
#include <hip/hip_runtime.h>

// int4 per-token fake-quant of K/V caches (B=4, S tokens, F=1024, fp32).
// One workgroup per (tensor, token): the 4096 floats sharing a scale live in
// one block -> single pass over memory (read once, write once), which is the
// only thing that matters at 23.3 TB/s. Data is staged global->LDS with the
// CDNA5 async tensor-path (global_load_async_to_lds_b128 / s_wait_asynccnt).

#ifndef USE_ASYNC_LDS
#define USE_ASYNC_LDS 1
#endif

typedef __attribute__((ext_vector_type(4))) float v4f;   // native vector: NT-store OK

namespace {
constexpr int   kB    = 4;     // batch
constexpr int   kF    = 1024;  // features
constexpr int   kTPB  = 256;   // 8 wave32s; 256 lanes * 4 floats = one row
constexpr float kQMAX = 7.0f;
}

__global__ __launch_bounds__(kTPB)
void kv_int4_fakequant_kernel(const float* __restrict__ kin,
                              const float* __restrict__ vin,
                              float* __restrict__ out,
                              int S)
{
    const int t   = blockIdx.x;        // token index
    const int sel = blockIdx.y;        // 0 = k, 1 = v
    const float* __restrict__ x = sel ? vin : kin;
    const size_t plane = (size_t)kB * (size_t)S * (size_t)kF;
    float* __restrict__ o = out + (size_t)sel * plane;

    const int    tid       = threadIdx.x;
    const size_t rowstride = (size_t)S * (size_t)kF;   // batch stride
    const size_t tok       = (size_t)t * (size_t)kF;   // token offset in a row
    const size_t lane_off  = (size_t)tid * 4;          // float4 per row per lane

    v4f d[kB];

#if USE_ASYNC_LDS
    __shared__ float stage[kB * kF];   // 16 KB staging tile
    // CDNA5 async copy: per-lane global b128 -> LDS, tracked by ASYNCcnt.
    // Flat-address low 32 bits == LDS byte offset (HW adds LDS_BASE).
    #pragma unroll
    for (int b = 0; b < kB; ++b) {
        const float* gp = x + (size_t)b * rowstride + tok + lane_off;
        unsigned     lp = (unsigned)(size_t)(&stage[b * kF + tid * 4]);
        asm volatile("global_load_async_to_lds_b128 %0, %1, off"
                     :: "v"(lp), "v"(gp) : "memory");
    }
    asm volatile("s_wait_asynccnt 0" ::: "memory");
    #pragma unroll
    for (int b = 0; b < kB; ++b)
        d[b] = *reinterpret_cast<const v4f*>(&stage[b * kF + tid * 4]);
#else
    #pragma unroll
    for (int b = 0; b < kB; ++b)
        d[b] = *reinterpret_cast<const v4f*>(x + (size_t)b * rowstride + tok + lane_off);
#endif

    // ---- per-token absmax: 16 regs -> wave32 -> block --------------------
    float m = 0.0f;
    #pragma unroll
    for (int b = 0; b < kB; ++b) {
        m = fmaxf(m, fabsf(d[b].x));
        m = fmaxf(m, fabsf(d[b].y));
        m = fmaxf(m, fabsf(d[b].z));
        m = fmaxf(m, fabsf(d[b].w));
    }
    #pragma unroll
    for (int off = 16; off > 0; off >>= 1)
        m = fmaxf(m, __shfl_xor(m, off, 32));

    __shared__ float wmax[kTPB / 32];
    if ((tid & 31) == 0) wmax[tid >> 5] = m;
    __syncthreads();
    float amax = wmax[0];
    #pragma unroll
    for (int w = 1; w < kTPB / 32; ++w) amax = fmaxf(amax, wmax[w]);

    // ---- quantize: q = clip(rint(x/scale), -8, 7) * scale ----------------
    const float scale = amax / kQMAX;          // one precise div per thread
    const bool  nz    = scale > 0.0f;
    const float safe  = nz ? scale : 1.0f;
    const float inv   = 1.0f / safe;           // per-element div -> multiply

    #pragma unroll
    for (int b = 0; b < kB; ++b) {
        v4f r;
        {
            float q;
            q = fminf(fmaxf(__builtin_rintf(d[b].x * inv), -8.0f), 7.0f) * safe;
            r.x = nz ? q : d[b].x;
            q = fminf(fmaxf(__builtin_rintf(d[b].y * inv), -8.0f), 7.0f) * safe;
            r.y = nz ? q : d[b].y;
            q = fminf(fmaxf(__builtin_rintf(d[b].z * inv), -8.0f), 7.0f) * safe;
            r.z = nz ? q : d[b].z;
            q = fminf(fmaxf(__builtin_rintf(d[b].w * inv), -8.0f), 7.0f) * safe;
            r.w = nz ? q : d[b].w;
        }
        // streamed once, never re-read -> non-temporal b128 store
        __builtin_nontemporal_store(
            r, reinterpret_cast<v4f*>(o + (size_t)b * rowstride + tok + lane_off));
    }
}

extern "C" void kernel_launch(void* const* d_in, const int* in_sizes, int n_in,
                              void* d_out, int out_size, void* d_ws, size_t ws_size,
                              hipStream_t stream) {
    const float* k   = (const float*)d_in[0];
    const float* v   = (const float*)d_in[1];
    float*       out = (float*)d_out;

    const int S = in_sizes[0] / (kB * kF);     // 4096 for the reference shapes
    dim3 grid((unsigned)S, 2u, 1u);
    kv_int4_fakequant_kernel<<<grid, kTPB, 0, stream>>>(k, v, out, S);
}